// TransE_Add_Time_90512140796654
// MI455X (gfx1250) — compile-verified
//
#include <hip/hip_runtime.h>

// TransE-with-time scoring for MI455X (gfx1250, wave32).
// One workgroup (8 waves) per triplet b:
//   score[b] = sum_e | l2n(h@M)[e] + r[e] - l2n(t@M)[e] + 1e-6 |,  M = Ts+Te
// Raw Ts/Te tiles staged to LDS with GLOBAL_LOAD_ASYNC_TO_LDS_B128 (ASYNCcnt),
// per-wave V_WMMA_F32_16X16X4_F32 computes [h;t] x M column-tile, K in 4 phases.

typedef __attribute__((ext_vector_type(2))) float v2f;
typedef __attribute__((ext_vector_type(8))) float v8f;
typedef __attribute__((ext_vector_type(4))) int   v4i;

#define TT_D 128
#define LDSTRIDE 144           // 128 + 16 pad: half-wave bank ranges disjoint
#define ROWS_PER_PHASE 32
#define NPHASES 4              // 4 * 32 = 128 = K
#define J_PER 4                // float4 chunks per thread per table per phase

#if __has_builtin(__builtin_amdgcn_global_load_async_to_lds_b128)
#define HAVE_ASYNC_LDS 1
#else
#define HAVE_ASYNC_LDS 0
#endif

__global__ __launch_bounds__(256) void transe_time_wmma_kernel(
    const int* __restrict__ triplet_idx,     // (B,5)
    const float* __restrict__ entity_emb,    // (500000,128)
    const float* __restrict__ relation_emb,  // (1000,128)
    const float* __restrict__ time_transfer, // (1000,128*128)
    float* __restrict__ out)                 // (B,)
{
    __shared__ float lds_Ts[ROWS_PER_PHASE * LDSTRIDE]; // 18432 B
    __shared__ float lds_Te[ROWS_PER_PHASE * LDSTRIDE]; // 18432 B
    __shared__ float lds_htz[3 * TT_D];                 // [h | t | zeros]
    __shared__ float lds_h2[TT_D];
    __shared__ float lds_t2[TT_D];

    const int b    = blockIdx.x;
    const int tid  = threadIdx.x;
    const int lane = tid & 31;
    const int wave = tid >> 5;     // 0..7 -> e-tile
    const int m    = lane & 15;    // row (A) / col (B,D) index inside tile
    const int half = lane >> 4;    // 0/1 -> K split within fragment

    const int i0 = triplet_idx[b * 5 + 0];
    const int i1 = triplet_idx[b * 5 + 1];
    const int i2 = triplet_idx[b * 5 + 2];
    const int i3 = triplet_idx[b * 5 + 3];
    const int i4 = triplet_idx[b * 5 + 4];

    // Stage h, t and a zero block contiguously: A-operand rows select by index.
    if (tid < TT_D) {
        lds_htz[tid]            = entity_emb[i0 * TT_D + tid];
        lds_htz[2 * TT_D + tid] = 0.0f;
    } else {
        lds_htz[tid] = entity_emb[i2 * TT_D + (tid - TT_D)]; // [128..255] = t
    }

    const float* Ts = time_transfer + (size_t)i3 * (TT_D * TT_D);
    const float* Te = time_transfer + (size_t)i4 * (TT_D * TT_D);

    // Per-lane constant fragment bases (no per-step selects -> no divergence).
    const int aoff    = (m == 0) ? 0 : ((m == 1) ? TT_D : 2 * TT_D);
    const int abase_i = aoff + 2 * half;                 // + rowbase + 4*kk
    const int bbase_i = 2 * half * LDSTRIDE + wave * 16 + m; // + 4*kk*LDSTRIDE

    v8f acc = {};

    for (int phase = 0; phase < NPHASES; ++phase) {
        __syncthreads();  // phase 0: h/t visible; later: prior readers done
        const int rowbase = phase * ROWS_PER_PHASE;

        // ---- stage 32 rows of raw Ts and Te into LDS (padded stride) ----
        #pragma unroll
        for (int j = 0; j < J_PER; ++j) {
            int o   = j * 1024 + tid * 4;   // float4-granular linear offset
            int kl  = o >> 7;               // local row 0..31
            int col = o & 127;
            int gidx = (rowbase + kl) * TT_D + col;
            int lidx = kl * LDSTRIDE + col;
#if HAVE_ASYNC_LDS
            __builtin_amdgcn_global_load_async_to_lds_b128(
                (__attribute__((address_space(1))) v4i*)(Ts + gidx),
                (__attribute__((address_space(3))) v4i*)(&lds_Ts[lidx]), 0, 0);
            __builtin_amdgcn_global_load_async_to_lds_b128(
                (__attribute__((address_space(1))) v4i*)(Te + gidx),
                (__attribute__((address_space(3))) v4i*)(&lds_Te[lidx]), 0, 0);
#else
            *(float4*)&lds_Ts[lidx] = *(const float4*)(Ts + gidx);
            *(float4*)&lds_Te[lidx] = *(const float4*)(Te + gidx);
#endif
        }
#if HAVE_ASYNC_LDS
#if __has_builtin(__builtin_amdgcn_s_wait_asynccnt)
        __builtin_amdgcn_s_wait_asynccnt(0);
#else
        asm volatile("s_wait_asynccnt 0x0" ::: "memory");
#endif
#endif
        __syncthreads();

        // ---- 8 K-steps of 4: WMMA f32 16x16x4, rows 0=h, 1=t, rest zero ----
        #pragma unroll
        for (int kk = 0; kk < 8; ++kk) {
            const int ka = abase_i + rowbase + kk * 4;   // h/t/zero (global k)
            v2f A;
            A.x = lds_htz[ka];
            A.y = lds_htz[ka + 1];
            const int kb = bbase_i + kk * 4 * LDSTRIDE;  // local rows
            v2f Bf;
            Bf.x = lds_Ts[kb]            + lds_Te[kb];
            Bf.y = lds_Ts[kb + LDSTRIDE] + lds_Te[kb + LDSTRIDE];
            acc = __builtin_amdgcn_wmma_f32_16x16x4_f32(
                false, A, false, Bf, (short)0, acc, false, false);
        }
    }

    // D layout: acc[r] lanes 0-15 = row r. Row 0 = h@M, row 1 = t@M.
    if (lane < 16) {
        lds_h2[wave * 16 + lane] = acc[0];
        lds_t2[wave * 16 + lane] = acc[1];
    }
    __syncthreads();

    if (wave == 0) {
        float sh = 0.f, st = 0.f;
        #pragma unroll
        for (int i = lane; i < TT_D; i += 32) {
            float a = lds_h2[i]; sh += a * a;
            float c = lds_t2[i]; st += c * c;
        }
        #pragma unroll
        for (int off = 16; off > 0; off >>= 1) {
            sh += __shfl_down(sh, off, 32);
            st += __shfl_down(st, off, 32);
        }
        sh = __shfl(sh, 0, 32);
        st = __shfl(st, 0, 32);
        float nh = fmaxf(sqrtf(sh), 1e-12f);
        float nt = fmaxf(sqrtf(st), 1e-12f);

        const float* r = relation_emb + (size_t)i1 * TT_D;
        float s = 0.f;
        #pragma unroll
        for (int i = lane; i < TT_D; i += 32) {
            float diff = lds_h2[i] / nh + r[i] - lds_t2[i] / nt + 1e-6f;
            s += fabsf(diff);   // P = 1 -> L1
        }
        #pragma unroll
        for (int off = 16; off > 0; off >>= 1)
            s += __shfl_down(s, off, 32);
        if (lane == 0) out[b] = s;
    }
}

extern "C" void kernel_launch(void* const* d_in, const int* in_sizes, int n_in,
                              void* d_out, int out_size, void* d_ws, size_t ws_size,
                              hipStream_t stream) {
    const int*   idx = (const int*)d_in[0];
    const float* ent = (const float*)d_in[1];
    const float* rel = (const float*)d_in[2];
    const float* tt  = (const float*)d_in[3];
    float* out = (float*)d_out;

    const int B = out_size;  // 8192 triplets, one workgroup each
    transe_time_wmma_kernel<<<B, 256, 0, stream>>>(idx, ent, rel, tt, out);
}